// _MultiHeadAttention_85753317032126
// MI455X (gfx1250) — compile-verified
//
#include <hip/hip_runtime.h>

// ---------------------------------------------------------------------------
// QKV projection for MHA on gfx1250 (MI455X).
//   Y[M=16384, N=1024] = X[M,K=1024] * W[K,N] + b, for X in {q,k,v},
//   then scatter-store as [B=4, H=16, S=4096, d=64].
// Roofline: 103 GFLOP vs ~400 MB of HBM traffic -> memory-bound (~17us @
// 23.3 TB/s). fp32 kept end-to-end using V_WMMA_F32_16X16X4_F32 (bit-exact
// precision class vs reference; matrix pipe engaged; no benefit to bf16
// down-conversion when HBM-bound).
// ---------------------------------------------------------------------------

typedef __attribute__((ext_vector_type(2))) float v2f;
typedef __attribute__((ext_vector_type(4))) float v4f;
typedef __attribute__((ext_vector_type(8))) float v8f;

#define DMODEL 1024
#define BM 128
#define BN 64
#define KC 32
#define XS_STRIDE 36   // floats; even (8B A-frag loads), 36*4 = 144 ≡ 0 mod 16, bank-spread ok
#define WS_STRIDE 68   // floats; 68*4 = 272 ≡ 0 mod 16

__device__ __forceinline__ void copy16(const float* __restrict__ g, float* __restrict__ l) {
  *(v4f*)l = *(const v4f*)g;   // global_load_b128 + ds_store_b128
}

__global__ __launch_bounds__(256)
void qkv_proj_kernel(const float* __restrict__ X, const float* __restrict__ W,
                     const float* __restrict__ bias, float* __restrict__ out)
{
  __shared__ float Xs[BM * XS_STRIDE];
  __shared__ float Ws[KC * WS_STRIDE];

  const int tid  = threadIdx.x;
  const int lane = tid & 31;
  const int wave = tid >> 5;

  const int nBlk = blockIdx.x * BN;   // N window: 64 cols
  const int mBlk = blockIdx.y * BM;   // M window: 128 rows
  const int mW   = wave * 16;         // wave's 16-row strip inside the block tile

  v8f acc0 = {}; v8f acc1 = {}; v8f acc2 = {}; v8f acc3 = {};

  // cooperative-load addressing (X: 128x32 floats, 16 floats/thread)
  const int xRow  = tid >> 1;        // 0..127
  const int xCol0 = (tid & 1) * 16;  // 0 or 16

  for (int kt = 0; kt < DMODEL; kt += KC) {
    // ---- stage X tile [BM x KC] into LDS ----
    {
      const float* xg = X + (size_t)(mBlk + xRow) * DMODEL + kt + xCol0;
      float* xl = &Xs[xRow * XS_STRIDE + xCol0];
#pragma unroll
      for (int j = 0; j < 4; ++j) copy16(xg + 4 * j, xl + 4 * j);
    }
    // ---- stage W tile [KC x BN] into LDS ----
#pragma unroll
    for (int u = 0; u < 2; ++u) {
      const int unit = tid + u * 256;      // 0..511 float4 units
      const int wRow = unit >> 4;          // 0..31
      const int wCol = (unit & 15) * 4;    // 0..60
      copy16(W + (size_t)(kt + wRow) * DMODEL + nBlk + wCol,
             &Ws[wRow * WS_STRIDE + wCol]);
    }
    __syncthreads();

    // ---- compute: 8 K-steps of 4, four 16x16 N-tiles per wave ----
    const int ar = lane & 15;            // A: row M
    const int ak = (lane >> 4) * 2;      // A: lanes 16-31 hold K=2,3
    const int bk = lane >> 4;            // B: lanes 16-31 hold K row k+1 (and k+3)
    const int bn = lane & 15;            // B: column N
#pragma unroll
    for (int kk = 0; kk < KC; kk += 4) {
      v2f a = *(const v2f*)&Xs[(mW + ar) * XS_STRIDE + kk + ak];
      const int r0 = (kk + bk) * WS_STRIDE + bn;
      const int r2 = (kk + bk + 2) * WS_STRIDE + bn;
      v2f b0, b1, b2, b3;
      b0.x = Ws[r0 +  0]; b0.y = Ws[r2 +  0];
      b1.x = Ws[r0 + 16]; b1.y = Ws[r2 + 16];
      b2.x = Ws[r0 + 32]; b2.y = Ws[r2 + 32];
      b3.x = Ws[r0 + 48]; b3.y = Ws[r2 + 48];
      acc0 = __builtin_amdgcn_wmma_f32_16x16x4_f32(false, a, false, b0, (short)0, acc0, false, false);
      acc1 = __builtin_amdgcn_wmma_f32_16x16x4_f32(false, a, false, b1, (short)0, acc1, false, false);
      acc2 = __builtin_amdgcn_wmma_f32_16x16x4_f32(false, a, false, b2, (short)0, acc2, false, false);
      acc3 = __builtin_amdgcn_wmma_f32_16x16x4_f32(false, a, false, b3, (short)0, acc3, false, false);
    }
    __syncthreads();
  }

  // ---- bias add + fused [B,S,H,d] -> [B,H,S,d] scatter-store ----
  // C/D layout: VGPR i -> M = i (lanes 0-15) / M = i+8 (lanes 16-31), N = lane%16.
  const int nLane = lane & 15;
  const int mHalf = (lane >> 4) * 8;
#pragma unroll
  for (int t = 0; t < 4; ++t) {
    const v8f acc = (t == 0) ? acc0 : (t == 1) ? acc1 : (t == 2) ? acc2 : acc3;
    const int n = nBlk + t * 16 + nLane;
    const float bz = bias[n];
    const int h = n >> 6, d = n & 63;
#pragma unroll
    for (int i = 0; i < 8; ++i) {
      const int m  = mBlk + mW + mHalf + i;
      const int bb = m >> 12;          // batch (4096 rows each)
      const int s  = m & 4095;         // sequence position
      const size_t idx = (((size_t)(bb * 16 + h) * 4096 + s) * 64) + d;
      out[idx] = acc[i] + bz;
    }
  }
}

extern "C" void kernel_launch(void* const* d_in, const int* in_sizes, int n_in,
                              void* d_out, int out_size, void* d_ws, size_t ws_size,
                              hipStream_t stream) {
  (void)in_sizes; (void)n_in; (void)out_size; (void)d_ws; (void)ws_size;
  const float* q  = (const float*)d_in[0];
  const float* k  = (const float*)d_in[1];
  const float* v  = (const float*)d_in[2];
  const float* Wq = (const float*)d_in[3];
  const float* bq = (const float*)d_in[4];
  const float* Wk = (const float*)d_in[5];
  const float* bk = (const float*)d_in[6];
  const float* Wv = (const float*)d_in[7];
  const float* bv = (const float*)d_in[8];
  float* out = (float*)d_out;

  const size_t perOut = (size_t)4 * 16 * 4096 * 64;  // 16,777,216 elements
  dim3 grid(DMODEL / BN, 16384 / BM);                // (16, 128)
  dim3 block(256);
  qkv_proj_kernel<<<grid, block, 0, stream>>>(q, Wq, bq, out);
  qkv_proj_kernel<<<grid, block, 0, stream>>>(k, Wk, bk, out + perOut);
  qkv_proj_kernel<<<grid, block, 0, stream>>>(v, Wv, bv, out + 2 * perOut);
}